// UNetProto_79620103733313
// MI455X (gfx1250) — compile-verified
//
#include <hip/hip_runtime.h>
#include <math.h>

typedef float v2f __attribute__((ext_vector_type(2)));
typedef float v8f __attribute__((ext_vector_type(8)));

#define NPIX   401408
#define HWPix  12544
#define EPSI   20.0f        // 1/SK_EPS
#define LNEPS  1e-5f

// output offsets (floats)
#define OUT_NEAREST 0
#define OUT_LOGITS  1605632
#define OUT_PT      4816896
#define OUT_PROTOS  5218304

// workspace offsets (floats)
#define WS_PN     0          // 2048
#define WS_E      2048       // NPIX*2
#define WS_CORR   804864     // NPIX
#define WS_PART1  1206272    // G1*12
#define WS_CLS    1507328    // 16
#define WS_W      1507360    // 24 (3 slots x 8)
#define WS_TPART  1507392    // G3*8
#define WS_FPART  1515584    // G4*2048
#define WS_NPART  2039872    // G4*8
#define WS_F      2041920    // 2048
#define WS_NS     2043968    // 8

#define G1 25088             // NPIX/16
#define G3 1024
#define G4 256
#define ROWS_PER_B4 1568     // NPIX/G4

// ---------------------------------------------------------------- K0: normalize prototypes
// pn[j = m*4+k][d] = l2norm(prototypes[k][m][:])[d]
__global__ __launch_bounds__(256) void k0_protos(const float* __restrict__ protos,
                                                 float* __restrict__ pn)
{
  int wave = threadIdx.x >> 5, lane = threadIdx.x & 31;
  int m = wave >> 2, k = wave & 3;             // j = m*4+k
  const float* src = protos + (k * 2 + m) * 256;
  float v[8]; float ss = 0.f;
#pragma unroll
  for (int i = 0; i < 8; ++i) { v[i] = src[lane + 32 * i]; ss += v[i] * v[i]; }
  for (int mm = 16; mm >= 1; mm >>= 1) ss += __shfl_xor(ss, mm, 32);
  float inv = 1.0f / fmaxf(sqrtf(ss), 1e-12f);
  float* dst = pn + wave * 256;
#pragma unroll
  for (int i = 0; i < 8; ++i) dst[lane + 32 * i] = v[i] * inv;
}

// ---------------------------------------------------------------- K1: main WMMA pass
// One wave per 16-row tile: LN + L2-norm + sim GEMM (WMMA f32 16x16x4) + nearest LN/argmax.
__global__ __launch_bounds__(32) void k1_main(
    const float* __restrict__ xin, const int* __restrict__ label,
    const float* __restrict__ fg, const float* __restrict__ fb,
    const float* __restrict__ mg, const float* __restrict__ mb,
    const float* __restrict__ pn,
    float* __restrict__ nearest_out, float* __restrict__ logits_out,
    float* __restrict__ e_out, float* __restrict__ corr_out,
    float* __restrict__ part1)
{
  __shared__ float S[16 * 260];      // feat tile (padded stride, 16B aligned rows)
  __shared__ float P[16 * 256];      // normalized prototypes (rows 8..15 zero)
  __shared__ float simT[16 * 8];
  __shared__ float rscale[16];
  const int lane = threadIdx.x;
  const int base = blockIdx.x * 16;

  { // prototypes -> LDS (rows 0..7), zero rows 8..15 so B loads are unconditional
    const float4* p4 = (const float4*)pn;
    float4* q4 = (float4*)P;
#pragma unroll
    for (int i = 0; i < 16; ++i) q4[lane + 32 * i] = p4[lane + 32 * i];
    float4 z; z.x = 0.f; z.y = 0.f; z.z = 0.f; z.w = 0.f;
#pragma unroll
    for (int i = 16; i < 32; ++i) q4[lane + 32 * i] = z;
  }
  { // 16x256 feat tile -> LDS
    const float4* g4 = (const float4*)xin;
#pragma unroll
    for (int i = 0; i < 32; ++i) {
      int f = lane + 32 * i;                   // float4 slot 0..1023
      int r = f >> 6, c4 = f & 63;
      float4 v = g4[(size_t)(base + r) * 64 + c4];
      float* s = &S[r * 260 + c4 * 4];
      s[0] = v.x; s[1] = v.y; s[2] = v.z; s[3] = v.w;
    }
  }
  __syncthreads();

  const int r = lane & 15, h = lane >> 4;      // 2 lanes per row for LN
  float sum = 0.f, ssq = 0.f;
  for (int c = h * 128; c < h * 128 + 128; ++c) { float x = S[r * 260 + c]; sum += x; ssq += x * x; }
  sum += __shfl_xor(sum, 16, 32); ssq += __shfl_xor(ssq, 16, 32);
  float mean = sum * (1.0f / 256.0f);
  float var  = ssq * (1.0f / 256.0f) - mean * mean;
  float rstd = rsqrtf(var + LNEPS);
  float l2 = 0.f;
  for (int c = h * 128; c < h * 128 + 128; ++c) {
    float y = (S[r * 260 + c] - mean) * rstd * fg[c] + fb[c];
    S[r * 260 + c] = y; l2 += y * y;
  }
  l2 += __shfl_xor(l2, 16, 32);
  float invn = 1.0f / fmaxf(sqrtf(l2), 1e-12f);
  if (lane < 16) rscale[lane] = invn;
  __syncthreads();

  // WMMA: D[16 rows][16 cols(8 used)] += A[16x4 feat] x B[4x16 protos]
  // A layout: lane l, vgpr v -> A[M=l%16][K = v + 2*(l/16)]
  // B layout (mirrored): lane l, vgpr v -> B[K = v + 2*(l/16)][N=l%16]
  // Row L2 scale is applied to the accumulator AFTER the K-loop (GEMM is linear).
  const float2* S2 = (const float2*)S;
  const float2* P2 = (const float2*)P;
  const int abase = r * 130 + h;               // float2 index of S[r][2h]
  const int bbase = r * 128 + h;               // float2 index of P[r][2h]
  v8f acc = {};
#pragma unroll
  for (int kb = 0; kb < 256; kb += 4) {
    float2 av = S2[abase + (kb >> 1)];
    float2 bv = P2[bbase + (kb >> 1)];
    v2f a; a.x = av.x; a.y = av.y;
    v2f b; b.x = bv.x; b.y = bv.y;
    acc = __builtin_amdgcn_wmma_f32_16x16x4_f32(false, a, false, b, (short)0, acc, false, false);
  }
  // C/D layout: lane l, vgpr v -> D[M = v + 8*(l/16)][N = l%16]
  if (r < 8) {
#pragma unroll
    for (int v = 0; v < 8; ++v) {
      int row = v + 8 * h;
      float sv = acc[v] * rscale[row];
      logits_out[(size_t)(base + row) * 8 + r] = sv;
      simT[row * 8 + r] = sv;
    }
  }
  __syncthreads();

  float pe0[4] = {0,0,0,0}, pe1[4] = {0,0,0,0}, pc[4] = {0,0,0,0};
  if (lane < 16) {
    int n = base + lane;
    float nr[4]; float m4 = 0.f;
#pragma unroll
    for (int k = 0; k < 4; ++k) { nr[k] = fmaxf(simT[lane * 8 + k], simT[lane * 8 + 4 + k]); m4 += nr[k]; }
    m4 *= 0.25f;
    float v4 = 0.f;
#pragma unroll
    for (int k = 0; k < 4; ++k) { float d = nr[k] - m4; v4 += d * d; }
    v4 *= 0.25f;
    float rs = rsqrtf(v4 + LNEPS);
    float best = -INFINITY; int pred = 0;
    int b2 = n / HWPix, rem = n - b2 * HWPix;
#pragma unroll
    for (int k = 0; k < 4; ++k) {
      float y = (nr[k] - m4) * rs * mg[k] + mb[k];
      nearest_out[(size_t)b2 * (4 * HWPix) + (size_t)k * HWPix + rem] = y;
      if (y > best) { best = y; pred = k; }     // first-max tie-break like jnp.argmax
    }
    int lab = label[n];
    corr_out[n] = (lab == pred) ? 1.f : 0.f;
    float e0 = __expf(simT[lane * 8 + lab] * EPSI);
    float e1 = __expf(simT[lane * 8 + 4 + lab] * EPSI);
    e_out[2 * (size_t)n] = e0; e_out[2 * (size_t)n + 1] = e1;
#pragma unroll
    for (int c = 0; c < 4; ++c) {
      float is = (lab == c) ? 1.f : 0.f;
      pe0[c] += is * e0; pe1[c] += is * e1; pc[c] += is;
    }
  }
#pragma unroll
  for (int c = 0; c < 4; ++c) {
    float a = pe0[c], bb = pe1[c], cc = pc[c];
    for (int m = 16; m >= 1; m >>= 1) {
      a += __shfl_xor(a, m, 32); bb += __shfl_xor(bb, m, 32); cc += __shfl_xor(cc, m, 32);
    }
    if (lane == 0) {
      part1[(size_t)blockIdx.x * 12 + c * 3 + 0] = a;
      part1[(size_t)blockIdx.x * 12 + c * 3 + 1] = bb;
      part1[(size_t)blockIdx.x * 12 + c * 3 + 2] = cc;
    }
  }
}

// ---------------------------------------------------------------- K2: class stats -> w^(1)
__global__ __launch_bounds__(256) void k2_cls(const float* __restrict__ part1,
                                              float* __restrict__ cls, float* __restrict__ w0)
{
  __shared__ float red[256];
  __shared__ float tot[12];
  int t = threadIdx.x;
  for (int i = 0; i < 12; ++i) {
    float a = 0.f;
    for (int b = t; b < G1; b += 256) a += part1[(size_t)b * 12 + i];
    red[t] = a; __syncthreads();
    for (int s = 128; s >= 1; s >>= 1) { if (t < s) red[t] += red[t + s]; __syncthreads(); }
    if (t == 0) tot[i] = red[0];
    __syncthreads();
  }
  if (t == 0) {
    for (int c = 0; c < 4; ++c) {
      float E0 = tot[c * 3], E1 = tot[c * 3 + 1], Bn = tot[c * 3 + 2];
      float S = E0 + E1;
      float w = 1.0f / fmaxf(S, 1e-30f);
      cls[c * 4 + 0] = Bn;
      cls[c * 4 + 1] = fmaxf(Bn, 1.0f);
      w0[c * 2 + 0] = w / fmaxf(w * E0, 1e-30f) * 0.5f;
      w0[c * 2 + 1] = w / fmaxf(w * E1, 1e-30f) * 0.5f;
    }
  }
}

// ---------------------------------------------------------------- K3a: T_m(w) partials
__global__ __launch_bounds__(256) void k3a_T(const float* __restrict__ e, const int* __restrict__ label,
                                             const float* __restrict__ win, float* __restrict__ tpart)
{
  __shared__ float red[256];
  int t = threadIdx.x;
  float acc[8] = {0,0,0,0,0,0,0,0};
  float wl[8];
#pragma unroll
  for (int i = 0; i < 8; ++i) wl[i] = win[i];
  for (size_t n = (size_t)blockIdx.x * 256 + t; n < NPIX; n += (size_t)G3 * 256) {
    int lab = label[n];
    float e0 = e[2 * n], e1 = e[2 * n + 1];
    float w0o = 0.f, w1o = 0.f;
#pragma unroll
    for (int c = 0; c < 4; ++c) { float is = (lab == c) ? 1.f : 0.f; w0o += is * wl[c * 2]; w1o += is * wl[c * 2 + 1]; }
    float den = fmaxf(w0o * e0 + w1o * e1, 1e-30f);
    float a0 = e0 / den, a1 = e1 / den;
#pragma unroll
    for (int c = 0; c < 4; ++c) { float is = (lab == c) ? 1.f : 0.f; acc[c * 2] += is * a0; acc[c * 2 + 1] += is * a1; }
  }
  for (int i = 0; i < 8; ++i) {
    red[t] = acc[i]; __syncthreads();
    for (int s = 128; s >= 1; s >>= 1) { if (t < s) red[t] += red[t + s]; __syncthreads(); }
    if (t == 0) tpart[(size_t)blockIdx.x * 8 + i] = red[0];
    __syncthreads();
  }
}

// ---------------------------------------------------------------- K3b: reduce T, update w
__global__ __launch_bounds__(256) void k3b_w(const float* __restrict__ tpart, const float* __restrict__ cls,
                                             const float* __restrict__ win, float* __restrict__ wout)
{
  __shared__ float red[256]; __shared__ float T8[8];
  int t = threadIdx.x;
  for (int i = 0; i < 8; ++i) {
    float a = 0.f;
    for (int b = t; b < G3; b += 256) a += tpart[(size_t)b * 8 + i];
    red[t] = a; __syncthreads();
    for (int s = 128; s >= 1; s >>= 1) { if (t < s) red[t] += red[t + s]; __syncthreads(); }
    if (t == 0) T8[i] = red[0];
    __syncthreads();
  }
  if (t == 0) {
    for (int c = 0; c < 4; ++c) {
      float Bnp = cls[c * 4 + 1];
      for (int m = 0; m < 2; ++m) {
        float w = win[c * 2 + m];
        float rs = w * T8[c * 2 + m] / Bnp;
        wout[c * 2 + m] = w / fmaxf(rs, 1e-30f) * 0.5f;
      }
    }
  }
}

// ---------------------------------------------------------------- K4: idx, proto_target, f/n partials
__global__ __launch_bounds__(256) void k4_acc(
    const float* __restrict__ xin, const int* __restrict__ label,
    const float* __restrict__ fg, const float* __restrict__ fb,
    const float* __restrict__ corr, const float* __restrict__ e,
    const float* __restrict__ w3,
    float* __restrict__ pt_out, float* __restrict__ fpart, float* __restrict__ npart)
{
  __shared__ float facc[8 * 256];
  __shared__ float ncnt[8];
  int t = threadIdx.x, lane = t & 31, wave = t >> 5;
#pragma unroll
  for (int j = 0; j < 8; ++j) facc[j * 256 + t] = 0.f;
  if (t < 8) ncnt[t] = 0.f;
  __syncthreads();
  float g[8], bta[8];
#pragma unroll
  for (int j = 0; j < 8; ++j) { g[j] = fg[lane * 8 + j]; bta[j] = fb[lane * 8 + j]; }
  float wl[8];
#pragma unroll
  for (int i = 0; i < 8; ++i) wl[i] = w3[i];
  size_t base = (size_t)blockIdx.x * ROWS_PER_B4;
  for (int i = 0; i < ROWS_PER_B4 / 8; ++i) {
    size_t n = base + (size_t)i * 8 + wave;
    const float4* g4 = (const float4*)xin + n * 64 + lane * 2;
    float4 v0 = g4[0], v1 = g4[1];
    float x[8] = {v0.x, v0.y, v0.z, v0.w, v1.x, v1.y, v1.z, v1.w};
    float sum = 0.f, ssq = 0.f;
#pragma unroll
    for (int j = 0; j < 8; ++j) { sum += x[j]; ssq += x[j] * x[j]; }
    for (int m = 16; m >= 1; m >>= 1) { sum += __shfl_xor(sum, m, 32); ssq += __shfl_xor(ssq, m, 32); }
    float mean = sum * (1.f / 256.f), var = ssq * (1.f / 256.f) - mean * mean;
    float rstd = rsqrtf(var + LNEPS);
    float y[8]; float l2 = 0.f;
#pragma unroll
    for (int j = 0; j < 8; ++j) { y[j] = (x[j] - mean) * rstd * g[j] + bta[j]; l2 += y[j] * y[j]; }
    for (int m = 16; m >= 1; m >>= 1) l2 += __shfl_xor(l2, m, 32);
    float invn = 1.0f / fmaxf(sqrtf(l2), 1e-12f);
    int lab = label[n];
    float e0 = e[2 * n], e1 = e[2 * n + 1];
    float w0o = 0.f, w1o = 0.f;
#pragma unroll
    for (int c = 0; c < 4; ++c) { float is = (lab == c) ? 1.f : 0.f; w0o += is * wl[c * 2]; w1o += is * wl[c * 2 + 1]; }
    int idx = (w1o * e1 > w0o * e0) ? 1 : 0;
    if (lane == 0) pt_out[n] = (float)(idx + 2 * lab);
    if (corr[n] != 0.f) {
      int fj = lab * 2 + idx;
#pragma unroll
      for (int j = 0; j < 8; ++j) atomicAdd(&facc[fj * 256 + lane * 8 + j], y[j] * invn);
      if (lane == 0) atomicAdd(&ncnt[fj], 1.0f);
    }
  }
  __syncthreads();
#pragma unroll
  for (int j = 0; j < 8; ++j) fpart[(size_t)blockIdx.x * 2048 + j * 256 + t] = facc[j * 256 + t];
  if (t < 8) npart[(size_t)blockIdx.x * 8 + t] = ncnt[t];
}

// ---------------------------------------------------------------- K5: reduce f/n partials
__global__ __launch_bounds__(256) void k5_red(const float* __restrict__ fpart, const float* __restrict__ npart,
                                              float* __restrict__ f, float* __restrict__ ns)
{
  int t = threadIdx.x;
#pragma unroll
  for (int j = 0; j < 8; ++j) {
    int fi = j * 256 + t;
    float a = 0.f;
    for (int b = 0; b < G4; ++b) a += fpart[(size_t)b * 2048 + fi];
    f[fi] = a;
  }
  if (t < 8) {
    float a = 0.f;
    for (int b = 0; b < G4; ++b) a += npart[(size_t)b * 8 + t];
    ns[t] = a;
  }
}

// ---------------------------------------------------------------- K6: momentum update + final L2 norm
__global__ __launch_bounds__(256) void k6_upd(const float* __restrict__ pn, const float* __restrict__ f,
                                              const float* __restrict__ ns, float* __restrict__ out)
{
  int t = threadIdx.x, lane = t & 31, fj = t >> 5;   // fj = c*2+m (output row order)
  int c = fj >> 1, m = fj & 1, jpn = m * 4 + c;      // pn row index
  float fv[8]; float ss = 0.f;
#pragma unroll
  for (int i = 0; i < 8; ++i) { fv[i] = f[fj * 256 + lane + 32 * i]; ss += fv[i] * fv[i]; }
  for (int mm = 16; mm >= 1; mm >>= 1) ss += __shfl_xor(ss, mm, 32);
  float fin = 1.0f / fmaxf(sqrtf(ss), 1e-12f);
  float nm = ns[fj];
  float ntot = ns[c * 2] + ns[c * 2 + 1];
  bool cond = (ntot > 0.f) && (nm != 0.f);
  float upd[8]; float ss2 = 0.f;
#pragma unroll
  for (int i = 0; i < 8; ++i) {
    float p = pn[jpn * 256 + lane + 32 * i];
    upd[i] = cond ? (0.999f * p + 0.001f * fv[i] * fin) : p;
    ss2 += upd[i] * upd[i];
  }
  for (int mm = 16; mm >= 1; mm >>= 1) ss2 += __shfl_xor(ss2, mm, 32);
  float inv2 = 1.0f / fmaxf(sqrtf(ss2), 1e-12f);
#pragma unroll
  for (int i = 0; i < 8; ++i) out[fj * 256 + lane + 32 * i] = upd[i] * inv2;
}

// ----------------------------------------------------------------
extern "C" void kernel_launch(void* const* d_in, const int* in_sizes, int n_in,
                              void* d_out, int out_size, void* d_ws, size_t ws_size,
                              hipStream_t stream)
{
  (void)in_sizes; (void)n_in; (void)out_size; (void)ws_size;
  const float* xin    = (const float*)d_in[0];
  const int*   label  = (const int*)d_in[1];
  const float* protos = (const float*)d_in[2];
  const float* fg     = (const float*)d_in[3];
  const float* fb     = (const float*)d_in[4];
  const float* mg     = (const float*)d_in[5];
  const float* mb     = (const float*)d_in[6];
  float* out = (float*)d_out;
  float* ws  = (float*)d_ws;

  float* nearest_out = out + OUT_NEAREST;
  float* logits_out  = out + OUT_LOGITS;
  float* pt_out      = out + OUT_PT;
  float* protos_out  = out + OUT_PROTOS;

  float* pn    = ws + WS_PN;
  float* e     = ws + WS_E;
  float* corr  = ws + WS_CORR;
  float* part1 = ws + WS_PART1;
  float* cls   = ws + WS_CLS;
  float* w0    = ws + WS_W;
  float* w1    = ws + WS_W + 8;
  float* w2    = ws + WS_W + 16;
  float* tpart = ws + WS_TPART;
  float* fpart = ws + WS_FPART;
  float* npart = ws + WS_NPART;
  float* f     = ws + WS_F;
  float* ns    = ws + WS_NS;

  k0_protos<<<1, 256, 0, stream>>>(protos, pn);
  k1_main<<<G1, 32, 0, stream>>>(xin, label, fg, fb, mg, mb, pn,
                                 nearest_out, logits_out, e, corr, part1);
  k2_cls<<<1, 256, 0, stream>>>(part1, cls, w0);
  k3a_T<<<G3, 256, 0, stream>>>(e, label, w0, tpart);
  k3b_w<<<1, 256, 0, stream>>>(tpart, cls, w0, w1);
  k3a_T<<<G3, 256, 0, stream>>>(e, label, w1, tpart);
  k3b_w<<<1, 256, 0, stream>>>(tpart, cls, w1, w2);
  k4_acc<<<G4, 256, 0, stream>>>(xin, label, fg, fb, corr, e, w2, pt_out, fpart, npart);
  k5_red<<<1, 256, 0, stream>>>(fpart, npart, f, ns);
  k6_upd<<<1, 256, 0, stream>>>(pn, f, ns, protos_out);
}